// Quantizer_62277025792084
// MI455X (gfx1250) — compile-verified
//
#include <hip/hip_runtime.h>

typedef __attribute__((ext_vector_type(2))) float v2f;
typedef __attribute__((ext_vector_type(8))) float v8f;
typedef int v4i __attribute__((vector_size(4 * sizeof(int))));   // matches builtin's pointee

#define DIMS 128
#define TILE_C 128          // codes staged per chunk (64 KB)
#define LDS_STRIDE 132      // 128 + 4 pad -> 16 lanes hit distinct banks (4*L mod 64)

// ---- CDNA5 async global->LDS path (ASYNCcnt), guarded so we compile either way
#if defined(__has_builtin)
#if __has_builtin(__builtin_amdgcn_global_load_async_to_lds_b128) && \
    __has_builtin(__builtin_amdgcn_s_wait_asynccnt)
#define HAVE_ASYNC_LDS 1
#endif
#endif
#ifndef HAVE_ASYNC_LDS
#define HAVE_ASYNC_LDS 0
#endif

#if HAVE_ASYNC_LDS
#define ASYNC_FENCE() __builtin_amdgcn_s_wait_asynccnt(0)
#else
#define ASYNC_FENCE() ((void)0)
#endif

// ---------------------------------------------------------------- ||c||^2
__global__ __launch_bounds__(256) void csq_kernel(const float* __restrict__ codes,
                                                  float* __restrict__ csq, int C) {
    int c = blockIdx.x * 256 + threadIdx.x;
    if (c < C) {
        const float4* p = (const float4*)(codes + (size_t)c * DIMS);
        float s = 0.f;
#pragma unroll
        for (int i = 0; i < DIMS / 4; ++i) {
            float4 v = p[i];
            s += v.x * v.x + v.y * v.y + v.z * v.z + v.w * v.w;
        }
        csq[c] = s;
    }
}

// stage one 128-code chunk into an LDS buffer (async DMA if available)
__device__ __forceinline__ void stage_chunk(const float* __restrict__ codes,
                                            const float* __restrict__ csq,
                                            float* ldsbuf, float* ldscsq,
                                            int cb, int tid) {
    for (int i = tid; i < TILE_C * (DIMS / 4); i += 128) {
        int r  = i >> 5;
        int k4 = (i & 31) << 2;
        const float* gsrc = &codes[(size_t)(cb + r) * DIMS + k4];
        float*       ldst = &ldsbuf[r * LDS_STRIDE + k4];
#if HAVE_ASYNC_LDS
        // AS1/AS3-qualified v4i* convert implicitly to generic v4i* if needed
        __builtin_amdgcn_global_load_async_to_lds_b128(
            (__attribute__((address_space(1))) v4i*)gsrc,
            (__attribute__((address_space(3))) v4i*)ldst, 0, 0);
#else
        *(float4*)ldst = *(const float4*)gsrc;
#endif
    }
    if (tid < TILE_C) ldscsq[tid] = csq[cb + tid];
}

// ---------------------------------------------------------------- main VQ kernel
// 128 threads = 4 waves. Each wave owns TWO 16-row A tiles (32 rows of x): each B
// fragment from LDS feeds two independent WMMA chains (half the LDS bytes/WMMA,
// 2x accumulator ILP). Codes double-buffered in LDS; next chunk streams in via
// GLOBAL_LOAD_ASYNC_TO_LDS while the current chunk is consumed by the WMMA pipe.
__global__ __launch_bounds__(128) void vq_main(const float* __restrict__ x,
                                               const float* __restrict__ codes,
                                               const float* __restrict__ csq,
                                               float* __restrict__ outq,
                                               float* __restrict__ outidx,
                                               float* __restrict__ rowmin,
                                               int C) {
    __shared__ __align__(16) float lds_codes[2][TILE_C * LDS_STRIDE];
    __shared__ float lds_csq[2][TILE_C];

    const int tid  = threadIdx.x;
    const int lane = tid & 31;
    const int wave = tid >> 5;    // 0..3
    const int half = lane >> 4;   // which K-pair this lane supplies to WMMA
    const int lm   = lane & 15;   // M (for A) / N (for B/C/D)
    const int rowBase = blockIdx.x * 128 + wave * 32;

    // ---- Two A tiles in 16x4-f32 WMMA layout: a*[kk] covers K = 4*kk + 2*half + {0,1}
    v2f a0[32], a1[32];
    const float* xr0 = x + (size_t)(rowBase + lm) * DIMS + 2 * half;
    const float* xr1 = xr0 + 16 * DIMS;
#pragma unroll
    for (int kk = 0; kk < 32; ++kk) {
        float2 t0 = *(const float2*)(xr0 + kk * 4);
        float2 t1 = *(const float2*)(xr1 + kk * 4);
        a0[kk][0] = t0.x; a0[kk][1] = t0.y;
        a1[kk][0] = t1.x; a1[kk][1] = t1.y;
    }

    // ---- ||x_row||^2 (each lane has half the K values; partner is lane^16)
    float xsq0 = 0.f, xsq1 = 0.f;
#pragma unroll
    for (int kk = 0; kk < 32; ++kk) {
        xsq0 += a0[kk][0] * a0[kk][0] + a0[kk][1] * a0[kk][1];
        xsq1 += a1[kk][0] * a1[kk][0] + a1[kk][1] * a1[kk][1];
    }
    xsq0 += __shfl_xor(xsq0, 16, 32);  // lane L: ||x_{rowBase+(L&15)}||^2
    xsq1 += __shfl_xor(xsq1, 16, 32);  // lane L: ||x_{rowBase+16+(L&15)}||^2

    // running argmin of d' = ||c||^2 - 2 x.c  (||x||^2 is a per-row constant)
    float minv[2][8];
    int   mini[2][8];
#pragma unroll
    for (int j = 0; j < 8; ++j) {
        minv[0][j] = 3.4e38f; mini[0][j] = 0;
        minv[1][j] = 3.4e38f; mini[1][j] = 0;
    }

    // ---- prologue: stage chunk 0 into buffer 0
    stage_chunk(codes, csq, lds_codes[0], lds_csq[0], 0, tid);
    ASYNC_FENCE();
    __syncthreads();

    int buf = 0;
    for (int cb = 0; cb < C; cb += TILE_C, buf ^= 1) {
        // kick off async staging of the NEXT chunk into the other buffer
        // (safe: previous barrier guarantees all reads of that buffer completed)
        if (cb + TILE_C < C)
            stage_chunk(codes, csq, lds_codes[buf ^ 1], lds_csq[buf ^ 1],
                        cb + TILE_C, tid);

        // keep L2 one chunk ahead of the async engine
        if (cb + 2 * TILE_C < C) {
#pragma unroll
            for (int p = 0; p < 4; ++p)
                __builtin_prefetch(
                    &codes[(size_t)(cb + 2 * TILE_C) * DIMS + (size_t)(p * 128 + tid) * 32],
                    0, 1);
        }

        const float* lbuf = lds_codes[buf];
        const float* lcs  = lds_csq[buf];
#pragma unroll 1
        for (int nb = 0; nb < TILE_C / 16; ++nb) {
            const float* bb = &lbuf[(nb * 16 + lm) * LDS_STRIDE + 2 * half];
            v8f acc0 = {0.f, 0.f, 0.f, 0.f, 0.f, 0.f, 0.f, 0.f};
            v8f acc1 = {0.f, 0.f, 0.f, 0.f, 0.f, 0.f, 0.f, 0.f};
#pragma unroll
            for (int kk = 0; kk < 32; ++kk) {
                float2 tb = *(const float2*)(bb + kk * 4);
                v2f b;
                b[0] = tb.x;
                b[1] = tb.y;
                // same B operand feeds two independent accumulator chains
                acc0 = __builtin_amdgcn_wmma_f32_16x16x4_f32(
                    false, a0[kk], false, b, (short)0, acc0, false, false);
                acc1 = __builtin_amdgcn_wmma_f32_16x16x4_f32(
                    false, a1[kk], false, b, (short)0, acc1, false, false);
            }
            float cs  = lcs[nb * 16 + lm];
            int   col = cb + nb * 16 + lm;
#pragma unroll
            for (int j = 0; j < 8; ++j) {
                float d0 = __builtin_fmaf(-2.f, acc0[j], cs);
                float d1 = __builtin_fmaf(-2.f, acc1[j], cs);
                if (d0 < minv[0][j]) { minv[0][j] = d0; mini[0][j] = col; }
                if (d1 < minv[1][j]) { minv[1][j] = d1; mini[1][j] = col; }
            }
        }

        ASYNC_FENCE();     // our async writes to buf^1 have landed in LDS
        __syncthreads();   // everyone done reading buf; staged data visible
    }

    // ---- reduce (min,idx) across the 16-lane half holding the same rows
#pragma unroll
    for (int t = 0; t < 2; ++t) {
#pragma unroll
        for (int j = 0; j < 8; ++j) {
            float v   = minv[t][j];
            int   idx = mini[t][j];
#pragma unroll
            for (int m = 1; m < 16; m <<= 1) {
                float ov = __shfl_xor(v, m, 32);
                int   oi = __shfl_xor(idx, m, 32);
                if (ov < v || (ov == v && oi < idx)) { v = ov; idx = oi; }
            }
            minv[t][j] = v;
            mini[t][j] = idx;
        }
    }

    // ---- per-row outputs: indices (as float) + full min distance for the loss
#pragma unroll
    for (int t = 0; t < 2; ++t) {
#pragma unroll
        for (int j = 0; j < 8; ++j) {
            float xs = t ? xsq1 : xsq0;
            float xr = __shfl(xs, j + 8 * half, 32);  // uniform exec for the shuffle
            if (lm == j) {
                int rowG = rowBase + t * 16 + j + 8 * half;
                rowmin[rowG] = xr + minv[t][j];       // min_c ||x - c||^2
                outidx[rowG] = (float)mini[t][j];
            }
        }
    }

    // ---- gather quantized = codes[argmin]  (whole wave copies one row: 32 x float4)
#pragma unroll
    for (int t = 0; t < 2; ++t) {
#pragma unroll
        for (int j = 0; j < 8; ++j) {
            int i0 = __shfl(mini[t][j], j, 32);       // row rowBase + t*16 + j
            int i1 = __shfl(mini[t][j], j + 16, 32);  // row rowBase + t*16 + j + 8
            float4 q0 = *(const float4*)&codes[(size_t)i0 * DIMS + lane * 4];
            float4 q1 = *(const float4*)&codes[(size_t)i1 * DIMS + lane * 4];
            *(float4*)&outq[(size_t)(rowBase + t * 16 + j) * DIMS + lane * 4]     = q0;
            *(float4*)&outq[(size_t)(rowBase + t * 16 + j + 8) * DIMS + lane * 4] = q1;
        }
    }
}

// ---------------------------------------------------------------- loss reduction
__global__ __launch_bounds__(256) void loss_kernel(const float* __restrict__ rowmin,
                                                   float* __restrict__ outloss, int B) {
    __shared__ float sh[256];
    float s = 0.f;
    for (int i = threadIdx.x; i < B; i += 256) s += rowmin[i];
    sh[threadIdx.x] = s;
    __syncthreads();
    for (int off = 128; off > 0; off >>= 1) {
        if (threadIdx.x < off) sh[threadIdx.x] += sh[threadIdx.x + off];
        __syncthreads();
    }
    // loss_commit + BETA*loss_codebook, both forward-equal => 1.25 * mean(min dist)
    if (threadIdx.x == 0) *outloss = 1.25f * sh[0] / (float)B;
}

// ---------------------------------------------------------------- launcher
extern "C" void kernel_launch(void* const* d_in, const int* in_sizes, int n_in,
                              void* d_out, int out_size, void* d_ws, size_t ws_size,
                              hipStream_t stream) {
    const float* x     = (const float*)d_in[0];
    const float* codes = (const float*)d_in[1];
    const int B = in_sizes[0] / DIMS;   // 32768
    const int C = in_sizes[1] / DIMS;   // 4096

    float* outq    = (float*)d_out;                 // B*128 quantized (STE fwd == quantized)
    float* outidx  = outq + (size_t)B * DIMS;       // B indices (stored as float values)
    float* outloss = outidx + B;                    // 1 scalar

    float* csq    = (float*)d_ws;                   // C floats
    float* rowmin = csq + C;                        // B floats

    hipLaunchKernelGGL(csq_kernel, dim3((C + 255) / 256), dim3(256), 0, stream,
                       codes, csq, C);
    hipLaunchKernelGGL(vq_main, dim3(B / 128), dim3(128), 0, stream,
                       x, codes, csq, outq, outidx, rowmin, C);
    hipLaunchKernelGGL(loss_kernel, dim3(1), dim3(256), 0, stream,
                       rowmin, outloss, B);
}